// VA_68951404970016
// MI455X (gfx1250) — compile-verified
//
#include <hip/hip_runtime.h>

typedef __attribute__((ext_vector_type(2))) float v2f;
typedef __attribute__((ext_vector_type(8))) float v8f;

#define FEAT 64  // D == H == C == 64 in the reference

#if defined(__has_builtin)
#if __has_builtin(__builtin_amdgcn_wmma_f32_16x16x4_f32)
#define HAVE_WMMA_F32X4 1
#endif
#endif

// ---------------------------------------------------------------------------
// Fused per-edge kernel: one wave per edge.
//   a_e = <h[src_e], h[dst_e]>          (wave shfl-xor reduction)
//   neigh[dst_e] += a_e * h[src_e]      (global f32 atomics, L2-resident)
// Each lane owns features [2*lane, 2*lane+1] -> coalesced b64 loads.
// ---------------------------------------------------------------------------
__global__ __launch_bounds__(256) void va_edge_kernel(
    const float* __restrict__ h,
    const int* __restrict__ src,
    const int* __restrict__ dst,
    float* __restrict__ neigh,
    int E) {
  const int gtid = blockIdx.x * blockDim.x + threadIdx.x;
  const int e = gtid >> 5;          // wave-uniform edge id
  const int lane = threadIdx.x & 31;
  if (e >= E) return;

  // src/dst are identical across the wave; force scalar for SGPR addressing.
  const int s = __builtin_amdgcn_readfirstlane(src[e]);
  const int d = __builtin_amdgcn_readfirstlane(dst[e]);

  const float2* hs2 = (const float2*)(h + (size_t)s * FEAT);
  const float2* hd2 = (const float2*)(h + (size_t)d * FEAT);
  const float2 a2 = hs2[lane];
  const float2 b2 = hd2[lane];

  float p = a2.x * b2.x + a2.y * b2.y;
  // full-wave butterfly reduction; every lane ends with the total dot product
  p += __shfl_xor(p, 1);
  p += __shfl_xor(p, 2);
  p += __shfl_xor(p, 4);
  p += __shfl_xor(p, 8);
  p += __shfl_xor(p, 16);

  float* np = neigh + (size_t)d * FEAT + lane * 2;
  __hip_atomic_fetch_add(np,     p * a2.x, __ATOMIC_RELAXED, __HIP_MEMORY_SCOPE_AGENT);
  __hip_atomic_fetch_add(np + 1, p * a2.y, __ATOMIC_RELAXED, __HIP_MEMORY_SCOPE_AGENT);
}

// ---------------------------------------------------------------------------
// out = relu(A @ W^T), A:[N,64], W:[64,64] torch [out,in], out:[N,64].
// One wave computes a 16x16 output tile with V_WMMA_F32_16X16X4_F32,
// 16 WMMAs over K=64.
//
// f32 16x4 A layout (ISA 7.12.2): lanes 0-15 M=0..15 K={0,1}; lanes 16-31
// M=0..15 K={2,3}. B (4x16) mirrors with N across lanes. C/D: VGPR j holds
// row M=j (lanes 0-15) / M=j+8 (lanes 16-31), N = lane%16.
// ---------------------------------------------------------------------------
__global__ __launch_bounds__(256) void va_fc_relu_kernel(
    const float* __restrict__ A,
    const float* __restrict__ W,
    float* __restrict__ out,
    int N) {
  const int gtid = blockIdx.x * blockDim.x + threadIdx.x;
  const int wave = gtid >> 5;
  const int lane = threadIdx.x & 31;

  const int tilesPerRow = FEAT / 16;            // 4 column tiles
  const int m0 = (wave / tilesPerRow) * 16;
  const int n0 = (wave % tilesPerRow) * 16;
  if (m0 >= N) return;                          // wave-uniform exit

  const int half = lane >> 4;                   // 0: K+{0,1}  1: K+{2,3}
  const int l16 = lane & 15;
  const int koff = half * 2;

  int arowi = m0 + l16;
  if (arowi >= N) arowi = N - 1;                // safety clamp (N%16==0 here)
  const float* arow = A + (size_t)arowi * FEAT;
  const float* wrow = W + (size_t)(n0 + l16) * FEAT;  // B[k][n] = W[n][k]

#if HAVE_WMMA_F32X4
  v8f c = {};
#pragma unroll
  for (int k = 0; k < FEAT; k += 4) {
    v2f a, b;
    a.x = arow[k + koff];
    a.y = arow[k + koff + 1];
    b.x = wrow[k + koff];
    b.y = wrow[k + koff + 1];
    // 8 args: (neg_a, A, neg_b, B, c_mod, C, reuse_a, reuse_b)
    c = __builtin_amdgcn_wmma_f32_16x16x4_f32(
        false, a, false, b, (short)0, c, false, false);
  }
#else
  // VALU fallback producing the identical output mapping (no wmma emitted).
  v8f c = {};
#pragma unroll
  for (int j = 0; j < 8; ++j) {
    const float* ar = A + (size_t)(m0 + j + 8 * half) * FEAT;
    const float* wr = W + (size_t)(n0 + l16) * FEAT;
    float acc = 0.f;
#pragma unroll
    for (int k = 0; k < FEAT; ++k) acc += ar[k] * wr[k];
    c[j] = acc;
  }
#endif

  // store with fused ReLU: c[j] -> out[m0 + j + 8*half][n0 + l16]
  float* obase = out + (size_t)(m0 + 8 * half) * FEAT + n0 + l16;
#pragma unroll
  for (int j = 0; j < 8; ++j) {
    float v = c[j];
    obase[(size_t)j * FEAT] = v > 0.f ? v : 0.f;
  }
}

// ---------------------------------------------------------------------------
// Host-side launch. Inputs: x[N,64] f32, src[E] i32, dst[E] i32,
// W1[64,64] f32, W2[64,64] f32. Output: [N,64] f32.
// Workspace: one N*64 f32 buffer (neigh). Layer-1 activations are staged in
// d_out (read by the second edge kernel before fc-2 overwrites it).
// ---------------------------------------------------------------------------
extern "C" void kernel_launch(void* const* d_in, const int* in_sizes, int n_in,
                              void* d_out, int out_size, void* d_ws, size_t ws_size,
                              hipStream_t stream) {
  const float* x   = (const float*)d_in[0];
  const int*   src = (const int*)d_in[1];
  const int*   dst = (const int*)d_in[2];
  const float* W1  = (const float*)d_in[3];
  const float* W2  = (const float*)d_in[4];

  const int N = in_sizes[0] / FEAT;
  const int E = in_sizes[1];

  float* out   = (float*)d_out;
  float* neigh = (float*)d_ws;                  // N*64 floats
  const size_t neighBytes = (size_t)N * FEAT * sizeof(float);

  const int edgeBlocks = (E + 7) / 8;           // 8 waves (edges) / 256-block
  const int fcWaves    = ((N + 15) / 16) * (FEAT / 16);
  const int fcBlocks   = (fcWaves + 7) / 8;

  // ---- layer 0: x -> h1 (staged in d_out) ----
  hipMemsetAsync(neigh, 0, neighBytes, stream);
  va_edge_kernel<<<edgeBlocks, 256, 0, stream>>>(x, src, dst, neigh, E);
  va_fc_relu_kernel<<<fcBlocks, 256, 0, stream>>>(neigh, W1, out, N);

  // ---- layer 1: h1 -> out ----
  hipMemsetAsync(neigh, 0, neighBytes, stream);
  va_edge_kernel<<<edgeBlocks, 256, 0, stream>>>(out, src, dst, neigh, E);
  va_fc_relu_kernel<<<fcBlocks, 256, 0, stream>>>(neigh, W2, out, N);
}